// ZeroTSK_27496380629713
// MI455X (gfx1250) — compile-verified
//
#include <hip/hip_runtime.h>
#include <math.h>

// ZeroTSK: out = (Σ_k w(k) y[k]) / (Σ_k w(k)),  w separable over 10 base-6 digits.
// Memory-bound: one streaming pass over y (242 MB fp32) ≈ 10.4 us at 23.3 TB/s.
// WMMA f32 16x16x4 (fp32 in/accum) used as the reduction engine with B = ones.
// x4 unroll: 4 independent global_load_b128 (2 KB/wave) in flight per iteration
// to meet the bytes-in-flight requirement of the HBM roofline; NT hint since y
// (242 MB) exceeds L2 (192 MB) and has no reuse.

typedef __attribute__((ext_vector_type(2))) float v2f;
typedef __attribute__((ext_vector_type(4))) float v4f;
typedef __attribute__((ext_vector_type(8))) float v8f;

#define NN      10
#define PP      6
#define HALF    7776        // 6^5
#define TOTALE  60466176    // 6^10
#define CHUNKS  472392      // TOTALE / 128  (one wave-chunk = 32 lanes * float4)
#define GROUPS  118098      // CHUNKS / 4    (wave-group = 4 chunks = 2 KB/wave, exact)
#define BLOCKS  2048
#define TPB     256

__global__ void __launch_bounds__(TPB) tsk_main(const float* __restrict__ X,
                                                const float* __restrict__ a,
                                                const float* __restrict__ b,
                                                const float* __restrict__ y,
                                                float* __restrict__ partial)
{
    __shared__ float hrow[NN * PP];   // 60 membership values
    __shared__ float WH[HALF];        // product of digit-weights 0..4
    __shared__ float WL[HALF];        // product of digit-weights 5..9
    __shared__ float wpart[TPB / 32];

    const int tid = threadIdx.x;

    // Stage 1: 60 Gaussian memberships.
    if (tid < NN * PP) {
        const int i = tid / PP;
        const float d  = a[tid] - X[i];
        const float bb = b[tid];
        hrow[tid] = expf(-(d * d) / (2.0f * bb * bb));
    }
    __syncthreads();

    // Stage 2: separable weight tables (7776 entries each).
    for (int r = tid; r < HALF; r += TPB) {
        int t = r;
        const int d4 = t % 6; t /= 6;
        const int d3 = t % 6; t /= 6;
        const int d2 = t % 6; t /= 6;
        const int d1 = t % 6; t /= 6;
        const int d0 = t;                       // slowest digit
        WH[r] = hrow[0*PP + d0] * hrow[1*PP + d1] * hrow[2*PP + d2]
              * hrow[3*PP + d3] * hrow[4*PP + d4];
        WL[r] = hrow[5*PP + d0] * hrow[6*PP + d1] * hrow[7*PP + d2]
              * hrow[8*PP + d3] * hrow[9*PP + d4];
    }
    __syncthreads();

    // Stage 3: stream y. Loop predicate depends only on the wave-group index,
    // so EXEC stays all-1s inside the body (WMMA requirement).
    const int lane    = tid & 31;
    const int wave    = tid >> 5;
    const int gwave   = blockIdx.x * (TPB / 32) + wave;
    const int wstride = BLOCKS * (TPB / 32);

    v8f acc0 = {};
    v8f acc1 = {};
    const v2f bones = {1.0f, 1.0f};   // B = ones => each WMMA row-sums A into C

    for (int g = gwave; g < GROUPS; g += wstride) {
        const int base = g * 512 + lane * 4;    // wave covers 2048B contiguous

        // Issue 4 independent 128-bit non-temporal loads before any consumption.
        const v4f y0 = __builtin_nontemporal_load((const v4f*)(y + base));
        const v4f y1 = __builtin_nontemporal_load((const v4f*)(y + base + 128));
        const v4f y2 = __builtin_nontemporal_load((const v4f*)(y + base + 256));
        const v4f y3 = __builtin_nontemporal_load((const v4f*)(y + base + 384));

#pragma unroll
        for (int u = 0; u < 4; ++u) {
            const int bu = base + 128 * u;
            const int hi = bu / HALF;           // const-div -> mulhi
            const int lo = bu - hi * HALF;      // lo % 4 == 0 (7776 % 4 == 0)
            const float wh   = WH[hi];
            const v4f   wl4  = *(const v4f*)(WL + lo);   // ds_load_b128
            const v4f   yv   = (u == 0) ? y0 : (u == 1) ? y1 : (u == 2) ? y2 : y3;

            const v2f p01 = { wh * wl4.x * yv.x, wh * wl4.y * yv.y };
            const v2f p23 = { wh * wl4.z * yv.z, wh * wl4.w * yv.w };
            // Alternate accumulators: consecutive WMMAs are independent.
            acc0 = __builtin_amdgcn_wmma_f32_16x16x4_f32(false, p01, false, bones,
                                                         (short)0, acc0, false, false);
            acc1 = __builtin_amdgcn_wmma_f32_16x16x4_f32(false, p23, false, bones,
                                                         (short)0, acc1, false, false);
        }
    }

    // C/D layout: lane 0 holds C[0..7, 0] in VGPRs 0..7, lane 16 holds C[8..15, 0].
    const v8f acc = acc0 + acc1;
    float s = acc[0] + acc[1] + acc[2] + acc[3] + acc[4] + acc[5] + acc[6] + acc[7];
    const float tot = __shfl(s, 0, 32) + __shfl(s, 16, 32);
    if (lane == 0) wpart[wave] = tot;
    __syncthreads();

    if (tid == 0) {
        float p = 0.0f;
        for (int w = 0; w < TPB / 32; ++w) p += wpart[w];
        partial[blockIdx.x] = p;   // plain store: deterministic, no atomics
    }
}

__global__ void tsk_final(const float* __restrict__ X,
                          const float* __restrict__ a,
                          const float* __restrict__ b,
                          const float* __restrict__ partial,
                          float* __restrict__ out)
{
    if (threadIdx.x == 0 && blockIdx.x == 0) {
        float num = 0.0f;
        for (int i = 0; i < BLOCKS; ++i) num += partial[i];  // fixed order
        float den = 1.0f;
        for (int i = 0; i < NN; ++i) {
            float rs = 0.0f;
            for (int j = 0; j < PP; ++j) {
                const float d  = a[i * PP + j] - X[i];
                const float bb = b[i * PP + j];
                rs += expf(-(d * d) / (2.0f * bb * bb));
            }
            den *= rs;   // sum(h) factorizes exactly over the digit rows
        }
        out[0] = num / den;
    }
}

extern "C" void kernel_launch(void* const* d_in, const int* in_sizes, int n_in,
                              void* d_out, int out_size, void* d_ws, size_t ws_size,
                              hipStream_t stream)
{
    const float* X = (const float*)d_in[0];   // [10]
    const float* a = (const float*)d_in[1];   // [10,6]
    const float* b = (const float*)d_in[2];   // [10,6]
    const float* y = (const float*)d_in[3];   // [6^10] flat
    float* out     = (float*)d_out;           // scalar
    float* partial = (float*)d_ws;            // BLOCKS floats of scratch

    tsk_main<<<BLOCKS, TPB, 0, stream>>>(X, a, b, y, partial);
    tsk_final<<<1, 32, 0, stream>>>(X, a, b, partial, out);
}